// TreecrfLoss_mloss_44659069943947
// MI455X (gfx1250) — compile-verified
//
#include <hip/hip_runtime.h>
#include <hip/hip_bf16.h>

#define Bn 16
#define Ln 256
#define SP 257            // LDS row pitch (floats) to reduce bank conflicts
#define NEGV -1000000000.0f

typedef __attribute__((ext_vector_type(2))) float v2f;
typedef __attribute__((ext_vector_type(8))) float v8f;

__device__ __forceinline__ void lse_merge(float& m, float& S, float x) {
  // fold value x into running logsumexp state: value = m + log(S)
  if (x > m) { S = S * __expf(m - x) + 1.0f; m = x; }
  else       { S += __expf(x - m); }
}

// ---------------------------------------------------------------------------
// Kernel 1: CYK inside pass. One workgroup (256 thr = 8 wave32) per batch b.
// Full 256x256 fp32 chart resident in LDS (CDNA5: 320KB/WGP).
// Blocked (16x16) recurrence; inter-block split-point sums via WMMA f32.
// ---------------------------------------------------------------------------
__global__ void cyk_inside_kernel(const float* __restrict__ lg,
                                  const unsigned char* __restrict__ maskspan,
                                  float* __restrict__ s_ws,
                                  float* __restrict__ sT_ws,
                                  float* __restrict__ rowmax_ws,
                                  float* __restrict__ colmax_ws,
                                  float* __restrict__ logZ_ws,
                                  float* __restrict__ lens_ws) {
  extern __shared__ float sm[];
  float* chart = sm;                       // 256*SP
  float* cblk  = sm + Ln * SP;             // 8 waves * 16*16 GEMM partials
  float* scA   = cblk + 8 * 256;           // 8 waves * 16 row scales
  float* scB   = scA + 8 * 16;             // 8 waves * 16 col scales
  int*   icnt  = (int*)(scB + 8 * 16);     // sequence length counter

  const int b    = blockIdx.x;
  const int tid  = threadIdx.x;
  const int lane = tid & 31;
  const int wave = tid >> 5;
  const size_t base = (size_t)b * Ln * Ln;

  if (tid == 0) *icnt = 0;
  __syncthreads();
  atomicAdd(icnt, maskspan[base + tid] ? 1 : 0);      // row 0 of maskspan -> len
  chart[tid * SP + tid] = lg[base + tid * Ln + tid];  // width-0 base case
  __syncthreads();
  const int len = *icnt;

  // ---- d = 0: widths 1..15 inside the 16 diagonal blocks (16 thr/block) ----
  const int g = tid >> 4;
  const int r = tid & 15;
  for (int w = 1; w <= 15; ++w) {
    if (r + w <= 15) {
      int i = g * 16 + r, j = i + w;
      float m = -INFINITY, S = 0.0f;
      for (int k = i; k < j; ++k)
        lse_merge(m, S, chart[i * SP + k] + chart[(k + 1) * SP + j]);
      chart[i * SP + j] = lg[base + i * Ln + j] + m + __logf(S);
    }
    __syncthreads();
  }

  // ---- d >= 1: one 16x16 block per wave; WMMA over completed middle-k ----
  for (int d = 1; d <= 15; ++d) {
    for (int blk = wave; blk < 16 - d; blk += 8) {
      const int BI = blk, BJ = blk + d;
      const int RI = BI * 16, CJ = BJ * 16;
      const int kL = RI + 15;     // first middle split point
      const int kR = CJ - 1;      // last middle split point; kR-kL = 16*(d-1)
      const int half = lane >> 4, q = lane & 15;

      // per-row / per-col max scales over the middle-k range
      {
        float mx = -INFINITY;
        if (half == 0) {
          for (int k = kL; k <= kR; ++k) mx = fmaxf(mx, chart[(RI + q) * SP + k]);
          scA[wave * 16 + q] = mx;
        } else {
          for (int k = kL; k <= kR; ++k) mx = fmaxf(mx, chart[(k + 1) * SP + (CJ + q)]);
          scB[wave * 16 + q] = mx;
        }
      }
      asm volatile("s_wait_dscnt 0x0" ::: "memory");
      const float aS = scA[wave * 16 + q];
      const float bS = scB[wave * 16 + q];

      // GEMM in scaled-exp domain over [kL, kR-1]: exactly 4*(d-1) K=4 chunks,
      // no predication (k = kR is peeled into the tail below).
      v8f acc = {};
      {
        const float* arow = chart + (RI + q) * SP;
        for (int k0 = kL; k0 < kR; k0 += 4) {
          const int ka = k0 + (half ? 2 : 0);
          v2f af, bf;
          af.x = __expf(arow[ka]     - aS);
          af.y = __expf(arow[ka + 1] - aS);
          bf.x = __expf(chart[(ka + 1) * SP + (CJ + q)] - bS);
          bf.y = __expf(chart[(ka + 2) * SP + (CJ + q)] - bS);
          acc = __builtin_amdgcn_wmma_f32_16x16x4_f32(false, af, false, bf,
                                                      (short)0, acc, false, false);
        }
      }
      // spill C tile so the tail phase can address it per-entry
      float* cb = cblk + wave * 256;
#pragma unroll
      for (int rr = 0; rr < 8; ++rr)
        cb[(rr + half * 8) * 16 + q] = acc[rr];
      asm volatile("s_wait_dscnt 0x0" ::: "memory");

      // sequential tail over intra-block anti-diagonals t
      for (int t = 0; t <= 30; ++t) {
        const int j0  = (t > 15) ? (t - 15) : 0;
        const int cnt = 16 - ((t >= 15) ? (t - 15) : (15 - t));
        if (lane < cnt) {
          const int jl = j0 + lane;
          const int il = 15 - t + jl;
          const int i = RI + il, j = CJ + jl;
          float m = scA[wave * 16 + il] + scB[wave * 16 + jl];
          float S = cb[il * 16 + jl];
          lse_merge(m, S, chart[i * SP + kR] + chart[(kR + 1) * SP + j]); // peeled k
          for (int k = i; k < kL; ++k)          // left tail (rows below, done)
            lse_merge(m, S, chart[i * SP + k] + chart[(k + 1) * SP + j]);
          for (int k = CJ; k < j; ++k)          // right tail (cols left, done)
            lse_merge(m, S, chart[i * SP + k] + chart[(k + 1) * SP + j]);
          chart[i * SP + j] = lg[base + i * Ln + j] + m + __logf(S);
        }
        asm volatile("s_wait_dscnt 0x0" ::: "memory");
      }
    }
    __syncthreads();
  }

  // spill chart (+transpose, +row/col maxes) for the outside pass
  for (int off = tid; off < Ln * Ln; off += 256) {
    const int i = off >> 8, j = off & 255;
    s_ws[base + off] = (j >= i) ? chart[i * SP + j] : NEGV;
  }
  for (int off = tid; off < Ln * Ln; off += 256) {
    const int c = off >> 8, rr = off & 255;
    sT_ws[base + off] = (rr <= c) ? chart[rr * SP + c] : NEGV;
  }
  {
    float mx = NEGV;
    for (int j = tid; j < Ln; ++j) mx = fmaxf(mx, chart[tid * SP + j]);
    rowmax_ws[b * Ln + tid] = mx;
    float mc = NEGV;
    for (int ii = 0; ii <= tid; ++ii) mc = fmaxf(mc, chart[ii * SP + tid]);
    colmax_ws[b * Ln + tid] = mc;
  }
  if (tid == 0) {
    logZ_ws[b] = chart[0 * SP + (len - 1)];
    lens_ws[b] = (float)len;
  }
}

// ---------------------------------------------------------------------------
// Kernel 2: outside pass + span marginals, same WMMA band-blocking.
// beta_hat = beta + lg lives in LDS (init NEGV so invalid spans vanish in
// the scaled-exp domain). Left-parent family = exp(bh)·exp(s)^T over later
// block-cols; right-parent family = exp(s)^T·exp(bh) over earlier block-rows.
// ---------------------------------------------------------------------------
__global__ void cyk_outside_kernel(const float* __restrict__ lg,
                                   const float* __restrict__ s_ws,
                                   const float* __restrict__ sT_ws,
                                   const float* __restrict__ rowmax_ws,
                                   const float* __restrict__ colmax_ws,
                                   const float* __restrict__ logZ_ws,
                                   const float* __restrict__ lens_ws,
                                   float* __restrict__ marg_ws) {
  extern __shared__ float sm[];
  float* bh   = sm;                  // 256*SP
  float* bhRM = sm + Ln * SP;        // running row max of bh
  float* bhCM = bhRM + Ln;           // running col max of bh
  float* cbL  = bhCM + Ln;           // 8 waves * 256 left-lump C tiles
  float* cbR  = cbL + 8 * 256;       // 8 waves * 256 right-lump C tiles
  float* scLa = cbR + 8 * 256;       // 4 scale arrays, 8 waves * 16
  float* scLb = scLa + 8 * 16;
  float* scRa = scLb + 8 * 16;
  float* scRb = scRa + 8 * 16;

  const int b = blockIdx.x;
  const int tid = threadIdx.x;
  const int lane = tid & 31;
  const int wave = tid >> 5;
  const size_t base = (size_t)b * Ln * Ln;
  const int len = (int)lens_ws[b];
  const float logZ = logZ_ws[b];
  const float* s   = s_ws + base;
  const float* sT  = sT_ws + base;
  const float* rmx = rowmax_ws + b * Ln;
  const float* cmx = colmax_ws + b * Ln;

  for (int off = tid; off < Ln * SP; off += 256) bh[off] = NEGV;
  for (int off = tid; off < Ln; off += 256) { bhRM[off] = NEGV; bhCM[off] = NEGV; }
  __syncthreads();
  if (tid == 0) {
    const float v = lg[base + (len - 1)];   // beta(0,len-1) = 0
    bh[0 * SP + (len - 1)] = v;
    bhRM[0] = v; bhCM[len - 1] = v;
    marg_ws[base + (len - 1)] = 1.0f;
  }
  __syncthreads();

  for (int d = 15; d >= 0; --d) {
    for (int blk = wave; blk < 16 - d; blk += 8) {
      const int BI = blk, BJ = blk + d;
      const int RI = BI * 16, CJ = BJ * 16;
      if (CJ >= len) continue;               // wholly invalid block
      const int half = lane >> 4, q = lane & 15;
      const bool hasL = (BJ < 15);
      const bool hasR = (BI > 0);
      if (half == 0) {
        scLa[wave * 16 + q] = bhRM[RI + q];
        scLb[wave * 16 + q] = hasL ? rmx[CJ + q + 1] : NEGV;
      } else {
        scRa[wave * 16 + q] = hasR ? cmx[RI + q - 1] : NEGV;
        scRb[wave * 16 + q] = bhCM[CJ + q];
      }
      asm volatile("s_wait_dscnt 0x0" ::: "memory");
      const float la = scLa[wave * 16 + q], lb = scLb[wave * 16 + q];
      const float ra = scRa[wave * 16 + q], rb = scRb[wave * 16 + q];

      // left-parent GEMM: K = jp in [(BJ+1)*16, 255] (multiple-of-16 length)
      v8f accL = {};
      if (hasL) {
        const float* bhr  = bh + (RI + q) * SP;          // A: bh[i, jp]
        const float* srow = s + (size_t)(CJ + q + 1) * Ln; // B: s[j+1, jp]
        for (int k0 = (BJ + 1) * 16; k0 < Ln; k0 += 4) {
          const int ka = k0 + (half ? 2 : 0);
          v2f af, bf;
          af.x = __expf(bhr[ka]      - la);
          af.y = __expf(bhr[ka + 1]  - la);
          bf.x = __expf(srow[ka]     - lb);
          bf.y = __expf(srow[ka + 1] - lb);
          accL = __builtin_amdgcn_wmma_f32_16x16x4_f32(false, af, false, bf,
                                                       (short)0, accL, false, false);
        }
      }
      // right-parent GEMM: K = ip in [0, RI-1] (multiple-of-16 length)
      v8f accR = {};
      if (hasR) {
        const float* strow = sT + (size_t)(RI + q - 1) * Ln; // A: s[ip, i-1]
        const float* bhc   = bh + (CJ + q);                   // B: bh[ip, j]
        for (int k0 = 0; k0 < RI; k0 += 4) {
          const int ka = k0 + (half ? 2 : 0);
          v2f af, bf;
          af.x = __expf(strow[ka]        - ra);
          af.y = __expf(strow[ka + 1]    - ra);
          bf.x = __expf(bhc[ka * SP]       - rb);
          bf.y = __expf(bhc[(ka + 1) * SP] - rb);
          accR = __builtin_amdgcn_wmma_f32_16x16x4_f32(false, af, false, bf,
                                                       (short)0, accR, false, false);
        }
      }
      float* cl = cbL + wave * 256;
      float* cr = cbR + wave * 256;
#pragma unroll
      for (int rr = 0; rr < 8; ++rr) {
        cl[(rr + half * 8) * 16 + q] = accL[rr];
        cr[(rr + half * 8) * 16 + q] = accR[rr];
      }
      asm volatile("s_wait_dscnt 0x0" ::: "memory");

      // sequential tail; deps: same row to the right, same col above
      for (int t = 0; t <= 30; ++t) {
        const int il0 = (t > 15) ? (t - 15) : 0;
        const int cnt = 16 - ((t >= 15) ? (t - 15) : (15 - t));
        if (lane < cnt) {
          const int il = il0 + lane;
          const int jl = il + 15 - t;
          const int i = RI + il, j = CJ + jl;
          if ((i <= j) && (j < len) && !(i == 0 && j == len - 1)) {
            float m1 = scLa[wave * 16 + il] + scLb[wave * 16 + jl];
            float S1 = cl[il * 16 + jl];
            float m2 = scRa[wave * 16 + il] + scRb[wave * 16 + jl];
            float S2 = cr[il * 16 + jl];
            if (m2 > m1) { float tm = m1; m1 = m2; m2 = tm;
                           float ts = S1; S1 = S2; S2 = ts; }
            S1 += S2 * __expf(m2 - m1);
            const float* srow = s + (size_t)(j + 1) * Ln;
            for (int jp = j + 1; jp <= CJ + 15; ++jp)       // left tail
              lse_merge(m1, S1, bh[i * SP + jp] + srow[jp]);
            if (il > 0) {
              const float* strow = sT + (size_t)(i - 1) * Ln;
              for (int ip = RI; ip < i; ++ip)               // right tail
                lse_merge(m1, S1, bh[ip * SP + j] + strow[ip]);
            }
            const float beta = m1 + __logf(S1);
            marg_ws[base + (size_t)i * Ln + j] =
                __expf(s[(size_t)i * Ln + j] + beta - logZ);
            const float bhat = beta + lg[base + (size_t)i * Ln + j];
            bh[i * SP + j] = bhat;
            bhRM[i] = fmaxf(bhRM[i], bhat);
            bhCM[j] = fmaxf(bhCM[j], bhat);
          }
        }
        asm volatile("s_wait_dscnt 0x0" ::: "memory");
      }
    }
    __syncthreads();
  }
}

// ---------------------------------------------------------------------------
// Kernel 3: elementwise loss terms -> deterministic per-block partials
// ---------------------------------------------------------------------------
__global__ void loss_partial_kernel(const float* __restrict__ ph,
                                    const float* __restrict__ pt,
                                    const float* __restrict__ ph_arc,
                                    const int* __restrict__ spans_ind,
                                    const int* __restrict__ ph_ind,
                                    const int* __restrict__ pt_ind,
                                    const unsigned char* __restrict__ maskarc,
                                    const unsigned char* __restrict__ maskspan,
                                    const float* __restrict__ marg_ws,
                                    float* __restrict__ partials) {
  __shared__ float red[256 * 4];
  const int e = blockIdx.x * 256 + threadIdx.x;
  float a_span = 0.f, a_ph = 0.f, a_pt = 0.f, a_cnt = 0.f;
  if (maskspan[e]) {
    const float mg = marg_ws[e];
    const float x = ph_arc[e];
    const float p = 1.0f / (1.0f + __expf(-x));
    const float pm = p * mg;
    if (spans_ind[e] >= 2) a_span += __logf(pm);
    else                   a_span += log1pf(-pm);
  }
  if (maskarc[e]) {
    const float x1 = ph[e];
    a_ph += fmaxf(x1, 0.f) + log1pf(__expf(-fabsf(x1))) - x1 * (float)ph_ind[e];
    const float x2 = pt[e];
    a_pt += fmaxf(x2, 0.f) + log1pf(__expf(-fabsf(x2))) - x2 * (float)pt_ind[e];
    a_cnt += 1.0f;
  }
  const int tid = threadIdx.x;
  red[tid * 4 + 0] = a_span; red[tid * 4 + 1] = a_ph;
  red[tid * 4 + 2] = a_pt;   red[tid * 4 + 3] = a_cnt;
  __syncthreads();
  for (int st = 128; st > 0; st >>= 1) {
    if (tid < st)
      for (int c = 0; c < 4; ++c) red[tid * 4 + c] += red[(tid + st) * 4 + c];
    __syncthreads();
  }
  if (tid == 0)
    for (int c = 0; c < 4; ++c) partials[blockIdx.x * 4 + c] = red[c];
}

// ---------------------------------------------------------------------------
// Kernel 4: combine partials (fixed order) -> scalar loss
// ---------------------------------------------------------------------------
__global__ void loss_final_kernel(const float* __restrict__ partials,
                                  const float* __restrict__ lens_ws,
                                  float* __restrict__ out) {
  __shared__ float red[256 * 4];
  const int tid = threadIdx.x;
  float a[4] = {0.f, 0.f, 0.f, 0.f};
  for (int k = tid; k < 4096; k += 256)
    for (int c = 0; c < 4; ++c) a[c] += partials[k * 4 + c];
  for (int c = 0; c < 4; ++c) red[tid * 4 + c] = a[c];
  __syncthreads();
  for (int st = 128; st > 0; st >>= 1) {
    if (tid < st)
      for (int c = 0; c < 4; ++c) red[tid * 4 + c] += red[(tid + st) * 4 + c];
    __syncthreads();
  }
  if (tid == 0) {
    float lens_sum = 0.f;
    for (int bb = 0; bb < Bn; ++bb) lens_sum += lens_ws[bb];
    const float loss_spans = -red[0] / lens_sum;
    const float na = red[3];
    out[0] = 0.1f * loss_spans + 0.9f * (red[1] / na + red[2] / na);
  }
}

extern "C" void kernel_launch(void* const* d_in, const int* in_sizes, int n_in,
                              void* d_out, int out_size, void* d_ws, size_t ws_size,
                              hipStream_t stream) {
  (void)in_sizes; (void)n_in; (void)out_size; (void)ws_size;
  const float* span_logits = (const float*)d_in[0];
  const float* ph          = (const float*)d_in[1];
  const float* pt          = (const float*)d_in[2];
  const float* ph_arc      = (const float*)d_in[3];
  const int*   spans_ind   = (const int*)d_in[4];
  const int*   ph_ind      = (const int*)d_in[5];
  const int*   pt_ind      = (const int*)d_in[6];
  const unsigned char* maskarc  = (const unsigned char*)d_in[7];
  const unsigned char* maskspan = (const unsigned char*)d_in[8];

  float* ws        = (float*)d_ws;
  float* s_ws      = ws;                        // 16*256*256
  float* sT_ws     = s_ws + Bn * Ln * Ln;       // 16*256*256
  float* marg_ws   = sT_ws + Bn * Ln * Ln;      // 16*256*256
  float* rowmax_ws = marg_ws + Bn * Ln * Ln;    // 16*256
  float* colmax_ws = rowmax_ws + Bn * Ln;       // 16*256
  float* logZ_ws   = colmax_ws + Bn * Ln;       // 16
  float* lens_ws   = logZ_ws + Bn;              // 16
  float* partials  = lens_ws + Bn;              // 4096*4

  const size_t smemA = (size_t)(Ln * SP + 8 * 256 + 8 * 16 + 8 * 16 + 1) * sizeof(float);
  const size_t smemB = (size_t)(Ln * SP + 2 * Ln + 2 * 8 * 256 + 4 * 8 * 16) * sizeof(float);
  hipFuncSetAttribute((const void*)cyk_inside_kernel,
                      hipFuncAttributeMaxDynamicSharedMemorySize, (int)smemA);
  hipFuncSetAttribute((const void*)cyk_outside_kernel,
                      hipFuncAttributeMaxDynamicSharedMemorySize, (int)smemB);

  cyk_inside_kernel<<<Bn, 256, smemA, stream>>>(span_logits, maskspan,
                                                s_ws, sT_ws, rowmax_ws, colmax_ws,
                                                logZ_ws, lens_ws);
  cyk_outside_kernel<<<Bn, 256, smemB, stream>>>(span_logits, s_ws, sT_ws,
                                                 rowmax_ws, colmax_ws,
                                                 logZ_ws, lens_ws, marg_ws);
  loss_partial_kernel<<<4096, 256, 0, stream>>>(ph, pt, ph_arc, spans_ind,
                                                ph_ind, pt_ind, maskarc,
                                                maskspan, marg_ws, partials);
  loss_final_kernel<<<1, 256, 0, stream>>>(partials, lens_ws, (float*)d_out);
}